// toy_gnn_distributed_39943195852845
// MI455X (gfx1250) — compile-verified
//
#include <hip/hip_runtime.h>
#include <stdint.h>

#define F_DIM 32
#define TILE  32          // edges staged per wave per buffer (== wave32 lanes)
#define WAVES_PER_BLOCK 8 // blockDim = 256

// ---------------------------------------------------------------------------
// Zero the output (harness poisons d_out with 0xAA before timing).
// ---------------------------------------------------------------------------
__global__ __launch_bounds__(256) void zero_kernel(float* __restrict__ out, int n)
{
    int tid    = blockIdx.x * blockDim.x + threadIdx.x;
    int stride = gridDim.x * blockDim.x;
    int n4 = n >> 2;
    float4* out4 = reinterpret_cast<float4*>(out);
    for (int i = tid; i < n4; i += stride)
        out4[i] = float4{0.0f, 0.0f, 0.0f, 0.0f};
    for (int i = (n4 << 2) + tid; i < n; i += stride)
        out[i] = 0.0f;
}

// ---------------------------------------------------------------------------
// Edge-parallel scatter-add. One wave per 32-edge tile, lane == feature.
// Edge metadata (src,dst,ew) is double-buffered into LDS as packed int4
// slots via CDNA5 async global->LDS copies (ASYNCcnt), overlapping the
// gather/atomic work; consumption is one ds_load_b128 broadcast per edge.
// ---------------------------------------------------------------------------
__global__ __launch_bounds__(256) void toy_gnn_scatter_kernel(
    const float* __restrict__ x,        // [N, 32]
    const int*   __restrict__ src,      // [E]
    const int*   __restrict__ dst,      // [E]
    const float* __restrict__ ew,       // [E]
    const float* __restrict__ w_mp,     // [1]
    float*       __restrict__ out,      // [N, 32]
    int nEdges)
{
    // Packed per-edge slot: {src, dst, ew_bits, pad} -> 16B, ds_load_b128-able.
    __shared__ int4 sh_edge[2][WAVES_PER_BLOCK][TILE];

    const int lane       = threadIdx.x & 31;
    const int waveInBlk  = threadIdx.x >> 5;
    const int waveId     = blockIdx.x * WAVES_PER_BLOCK + waveInBlk;
    const int totalWaves = gridDim.x * WAVES_PER_BLOCK;
    const int nTiles     = (nEdges + TILE - 1) / TILE;

    const float wm = w_mp[0];
    const uint32_t laneByte = (uint32_t)lane << 2;   // lane*4 bytes

    // Generic->LDS offset: low 32 bits of a generic LDS pointer are the
    // workgroup-relative LDS byte offset, which is exactly what the async
    // copy engine adds LDS_BASE to.
    #define LDS_OFF(p) ((uint32_t)(uintptr_t)(p))

    // Stage 32 edges (one per lane) into LDS buffer `buf` with async copies,
    // scattered into the interleaved int4 slots.
    auto stage = [&](int tile, int buf) {
        long idx = (long)tile * TILE + lane;
        if (idx >= nEdges) idx = nEdges - 1;   // clamp: harmless re-read
        uint32_t slot = LDS_OFF(&sh_edge[buf][waveInBlk][lane]);
        uint64_t gs = (uint64_t)(uintptr_t)(src + idx);
        uint64_t gd = (uint64_t)(uintptr_t)(dst + idx);
        uint64_t gw = (uint64_t)(uintptr_t)(ew  + idx);
        asm volatile("global_load_async_to_lds_b32 %0, %1, off"
                     :: "v"(slot), "v"(gs) : "memory");
        asm volatile("global_load_async_to_lds_b32 %0, %1, off offset:4"
                     :: "v"(slot), "v"(gd) : "memory");
        asm volatile("global_load_async_to_lds_b32 %0, %1, off offset:8"
                     :: "v"(slot), "v"(gw) : "memory");
    };

    int t = waveId;
    int buf = 0;
    if (t < nTiles) stage(t, 0);

    for (; t < nTiles; t += totalWaves) {
        int tn = t + totalWaves;
        if (tn < nTiles) {
            stage(tn, buf ^ 1);
            // 6 async ops outstanding; in-order completion => <=3 means the
            // current buffer's 3 copies have landed in LDS.
            asm volatile("s_wait_asynccnt 3" ::: "memory");
        } else {
            asm volatile("s_wait_asynccnt 0" ::: "memory");
        }

        long base = (long)t * TILE;
        int  rem  = nEdges - (int)base;
        int  m    = rem < TILE ? rem : TILE;

        // Prefetch all 32 gather rows of this tile in one instruction
        // (lane i prefetches the x-row of tile-edge i).
        {
            int sl = sh_edge[buf][waveInBlk][lane].x;
            __builtin_prefetch(&x[(size_t)sl * F_DIM], 0, 3);
        }

        for (int j = 0; j < m; ++j) {
            int4 e = sh_edge[buf][waveInBlk][j];     // one broadcast ds_load_b128
            int   s = e.x;
            float w = __int_as_float(e.z) * wm;

            float v = x[(size_t)s * F_DIM + lane] * w;   // coalesced 128B gather

            // Native f32 atomic add at L2, device scope, SADDR form:
            // uniform base in SGPRs + 32-bit per-lane byte offset.
            uint32_t ooff = ((uint32_t)e.y << 7) | laneByte;   // (d*32 + lane)*4
            asm volatile("global_atomic_add_f32 %0, %1, %2 scope:SCOPE_DEV"
                         :: "v"(ooff), "v"(v), "s"(out) : "memory");
        }
        buf ^= 1;
    }
    #undef LDS_OFF
}

// ---------------------------------------------------------------------------
// Harness entry point.
// Inputs (setup_inputs order): x [N*32] f32, edge_index [2*E] int,
// edge_weight [E] f32, w_mp [1] f32. Output: [N*32] f32.
// ---------------------------------------------------------------------------
extern "C" void kernel_launch(void* const* d_in, const int* in_sizes, int n_in,
                              void* d_out, int out_size, void* d_ws, size_t ws_size,
                              hipStream_t stream)
{
    const float* x    = (const float*)d_in[0];
    const int*   eidx = (const int*)  d_in[1];
    const float* ew   = (const float*)d_in[2];
    const float* wmp  = (const float*)d_in[3];
    float*       out  = (float*)d_out;

    const int E   = in_sizes[2];      // edge_weight element count
    const int* src = eidx;            // row 0 of [2, E]
    const int* dst = eidx + E;        // row 1 of [2, E]

    zero_kernel<<<1024, 256, 0, stream>>>(out, out_size);

    // 1024 blocks * 8 waves = 8192 tile-pipelines over 50k tiles (~6 each),
    // enough to fill the chip while keeping the async double-buffer busy.
    toy_gnn_scatter_kernel<<<1024, 256, 0, stream>>>(x, src, dst, ew, wmp, out, E);
}